// PatchletsExtractorBidirectional_83743272337983
// MI455X (gfx1250) — compile-verified
//
#include <hip/hip_runtime.h>
#include <hip/hip_bf16.h>
#include <stdint.h>

// ---------------------------------------------------------------------------
// PatchletsExtractorBidirectional for MI455X (gfx1250, wave32)
//
// Shapes: point_seq [B=4, T=32, N=1024, D=3], K=16, RADIUS=0.2
//
// Design:
//  * Per-query chains: grid = B * (8 fwd + 4 bwd) workgroups of 128 threads,
//    each thread owns one query's (x,y,z) carry in registers across T steps.
//  * Per step: frame staged to LDS planar (px/py/pz/||p||^2) with
//    global_load_async_to_lds_b32 (+ s_wait_asynccnt).
//  * Distances: d2 = ||q||^2 + ||p||^2 - 2 q.p via V_WMMA_F32_16X16X4_F32
//    (A = -2q as 16x4 with K=3 padded to 4; C preloaded with the norms).
//  * Top-16: one lane per query, register-resident sorted list, unrolled
//    bubble insertion (no dynamic indexing -> no scratch).
//  * Pass 2 kernel fills norm_pp and pf[...,3:6] from pp (anchor at t=0,k=0).
// ---------------------------------------------------------------------------

#define B_ 4
#define T_ 32
#define N_ 1024
#define K_ 16
#define R2_ 0.04f

typedef float v2f __attribute__((ext_vector_type(2)));
typedef float v8f __attribute__((ext_vector_type(8)));

__device__ __forceinline__ unsigned lds_off_u32(const void* p) {
  // generic -> LDS(addrspace 3) cast; AS3 pointers are 32-bit byte offsets
  return (unsigned)(uintptr_t)(__attribute__((address_space(3))) const void*)p;
}

__device__ __forceinline__ void async_b32(void* lds, const void* gptr) {
  unsigned lo = lds_off_u32(lds);
  asm volatile("global_load_async_to_lds_b32 %0, %1, off"
               :: "v"(lo), "v"(gptr) : "memory");
}

__global__ __launch_bounds__(128) void knn_chain_kernel(
    const float* __restrict__ x, float* __restrict__ out)
{
  __shared__ float sPx[N_], sPy[N_], sPz[N_], sDd[N_];   // 16 KB planar frame
  __shared__ float sTile[4][32 * 16];                    // 8 KB d2 tiles (per wave)

  const int tid  = threadIdx.x;
  const int lane = tid & 31;
  const int wv   = tid >> 5;
  const int lh   = lane & 15;
  const int hi   = lane >> 4;          // 0 for lanes 0-15, 1 for 16-31

  const int bidx = blockIdx.x;         // 0..47
  const int b    = bidx / 12;
  const int r    = bidx % 12;
  const int dir  = (r >= 8) ? 1 : 0;   // 8 fwd WGs (1024 q), 4 bwd WGs (512 q)
  const int wq   = dir ? (r - 8) : r;
  const int qn   = wq * 128 + tid;     // this lane's query index

  const size_t S1 = (size_t)B_ * T_ * N_ * K_;
  float* o_idx1 = out;
  float* o_dist = out + S1;
  float* o_idx2 = out + 2 * S1;
  float* o_pp   = out + 3 * S1;        // [B,T,N,K,3]
  float* o_pf   = out + 6 * S1;        // [B,T,N,K,6]
  float* o_outx = out + 15 * S1;       // [B,T,N,3]

  // initial carry: fwd -> frame 0, bwd -> frame T-1
  const float* q0 = x + (((size_t)b * T_ + (dir ? (T_ - 1) : 0)) * N_ + qn) * 3;
  float qx = q0[0], qy = q0[1], qz = q0[2];

  for (int t = 0; t < T_; ++t) {
    const int fN = dir ? (T_ - 1 - t) : t;                       // x_next frame
    const int fP = dir ? ((t == 0) ? (T_ - 1) : (T_ - t))        // f_prev frame
                       : ((t == 0) ? 0 : (t - 1));
    const int to = dir ? (T_ - 1 - t) : t;                       // output time

    // ---- stage frame fN into LDS (planar) via async global->LDS DMA ----
    asm volatile("s_wait_dscnt 0" ::: "memory"); // drain our LDS reads
    __syncthreads();                             // safe to overwrite LDS
    const float* fb = x + ((size_t)b * T_ + fN) * N_ * 3;
    for (int i = tid; i < N_; i += 128) {
      async_b32(&sPx[i], fb + (size_t)i * 3 + 0);
      async_b32(&sPy[i], fb + (size_t)i * 3 + 1);
      async_b32(&sPz[i], fb + (size_t)i * 3 + 2);
    }
    asm volatile("s_wait_asynccnt 0" ::: "memory");
    __syncthreads();
    for (int i = tid; i < N_; i += 128) {
      float px = sPx[i], py = sPy[i], pz = sPz[i];
      sDd[i] = px * px + py * py + pz * pz;
    }
    __syncthreads();

    // ---- per-step A fragments (ISA 16x4 f32 A layout) + qq broadcasts ----
    float qq  = qx * qx + qy * qy + qz * qz;
    float s0x = __shfl(qx, lh, 32),      s0y = __shfl(qy, lh, 32),      s0z = __shfl(qz, lh, 32);
    float s1x = __shfl(qx, lh + 16, 32), s1y = __shfl(qy, lh + 16, 32), s1z = __shfl(qz, lh + 16, 32);
    v2f a0, a1;                                  // A pre-scaled by -2
    a0.x = (lane < 16) ? (-2.f * s0x) : (-2.f * s0z);  // K0=x | K2=z
    a0.y = (lane < 16) ? (-2.f * s0y) : 0.f;           // K1=y | K3=0
    a1.x = (lane < 16) ? (-2.f * s1x) : (-2.f * s1z);
    a1.y = (lane < 16) ? (-2.f * s1y) : 0.f;
    float cq0[8], cq1[8];                        // C-layout qq (M = v + 8*hi)
#pragma unroll
    for (int v = 0; v < 8; ++v) {
      cq0[v] = __shfl(qq, v + 8 * hi, 32);
      cq1[v] = __shfl(qq, 16 + v + 8 * hi, 32);
    }

    float bd[K_]; int bi[K_];                    // sorted ascending top-K
#pragma unroll
    for (int k = 0; k < K_; ++k) { bd[k] = 3.4e38f; bi[k] = 0; }

    float* tb = &sTile[wv][0];

    // ---- 64 candidate tiles of 16 ----
    for (int ct = 0; ct < N_ / 16; ++ct) {
      const int ci = ct * 16 + lh;
      float bx = sPx[ci], by = sPy[ci], bz = sPz[ci];
      float ddv = sDd[ci];
      v2f bf;                                    // B 4x16 layout: V0:{K0|K2} V1:{K1|K3}
      bf.x = (lane < 16) ? bx : bz;
      bf.y = (lane < 16) ? by : 0.f;
      v8f c0, c1;
#pragma unroll
      for (int v = 0; v < 8; ++v) { c0[v] = cq0[v] + ddv; c1[v] = cq1[v] + ddv; }

      // d2 = (||q||^2 + ||p||^2) + (-2q).p
      v8f d0 = __builtin_amdgcn_wmma_f32_16x16x4_f32(false, a0, false, bf, (short)0, c0, false, false);
      v8f d1 = __builtin_amdgcn_wmma_f32_16x16x4_f32(false, a1, false, bf, (short)0, c1, false, false);

#pragma unroll
      for (int v = 0; v < 8; ++v) {              // C layout: M = v + 8*hi, N = lh
        tb[(v + 8 * hi) * 16 + lh]        = d0[v];
        tb[(16 + v + 8 * hi) * 16 + lh]   = d1[v];
      }
      asm volatile("s_wait_dscnt 0" ::: "memory"); // wave-internal LDS RAW fence

      // one lane per query: scan 16 candidates, unrolled insertion
#pragma unroll
      for (int j = 0; j < 16; ++j) {
        float d = tb[lane * 16 + j];
        int cidx = ct * 16 + j;
        if (d < bd[K_ - 1]) {
          bd[K_ - 1] = d; bi[K_ - 1] = cidx;
#pragma unroll
          for (int s = K_ - 1; s > 0; --s) {
            if (bd[s] < bd[s - 1]) {
              float td = bd[s]; bd[s] = bd[s - 1]; bd[s - 1] = td;
              int   ti = bi[s]; bi[s] = bi[s - 1]; bi[s - 1] = ti;
            }
          }
        }
      }
      asm volatile("" ::: "memory");
    }

    // ---- radius mask (dist = max(d2,0); >r^2 -> idx[0], 0) ----
    const int i0 = bi[0];
#pragma unroll
    for (int k = 0; k < K_; ++k) {
      float dk = fmaxf(bd[k], 0.f);
      int   ik = bi[k];
      if (dk > R2_) { ik = i0; dk = 0.f; }
      bd[k] = dk; bi[k] = ik;
    }

    const float nx = sPx[i0], ny = sPy[i0], nz = sPz[i0];  // x_new = pp[...,0,:]

    // ---- writes (identity stand-in for jax.random.permutation selection) ----
    if (qn < 512) {
      const int on = dir * 512 + qn;
      const size_t row = (((size_t)b * T_ + to) * N_ + on) * K_;
      const float* fp = x + ((size_t)b * T_ + fP) * N_ * 3;
#pragma unroll
      for (int k = 0; k < K_; ++k) {
        const int ik = bi[k];
        o_idx1[row + k] = (float)ik;
        o_idx2[row + k] = (float)ik;
        o_dist[row + k] = bd[k];
        o_pp[(row + k) * 3 + 0] = sPx[ik];
        o_pp[(row + k) * 3 + 1] = sPy[ik];
        o_pp[(row + k) * 3 + 2] = sPz[ik];
        o_pf[(row + k) * 6 + 0] = fp[(size_t)ik * 3 + 0];
        o_pf[(row + k) * 6 + 1] = fp[(size_t)ik * 3 + 1];
        o_pf[(row + k) * 6 + 2] = fp[(size_t)ik * 3 + 2];
      }
    }
    if (dir == 0) {
      const size_t orow = (((size_t)b * T_ + to) * N_ + qn) * 3;
      o_outx[orow + 0] = nx; o_outx[orow + 1] = ny; o_outx[orow + 2] = nz;
    }
    qx = nx; qy = ny; qz = nz;                   // advance the chain
  }
}

// Pass 2: norm_pp = pp - anchor(pp[b,0,n,0,:]); pf[...,3:6] = norm_pp
__global__ __launch_bounds__(256) void finalize_kernel(float* __restrict__ out)
{
  const size_t S1 = (size_t)B_ * T_ * N_ * K_;
  const float* pp  = out + 3 * S1;
  float*       pf  = out + 6 * S1;
  float*       npp = out + 12 * S1;
  size_t gid = (size_t)blockIdx.x * blockDim.x + threadIdx.x;
  if (gid >= S1) return;
  size_t nk = gid % ((size_t)N_ * K_);
  size_t bt = gid / ((size_t)N_ * K_);
  size_t bb = bt / T_;
  size_t n  = nk / K_;
  size_t aix = ((bb * T_ + 0) * N_ + n) * K_;    // (b, t=0, n, k=0)
  float ax = pp[aix * 3 + 0], ay = pp[aix * 3 + 1], az = pp[aix * 3 + 2];
  float v0 = pp[gid * 3 + 0] - ax;
  float v1 = pp[gid * 3 + 1] - ay;
  float v2 = pp[gid * 3 + 2] - az;
  npp[gid * 3 + 0] = v0; npp[gid * 3 + 1] = v1; npp[gid * 3 + 2] = v2;
  pf[gid * 6 + 3] = v0;  pf[gid * 6 + 4] = v1;  pf[gid * 6 + 5] = v2;
}

extern "C" void kernel_launch(void* const* d_in, const int* in_sizes, int n_in,
                              void* d_out, int out_size, void* d_ws, size_t ws_size,
                              hipStream_t stream) {
  (void)in_sizes; (void)n_in; (void)out_size; (void)d_ws; (void)ws_size;
  const float* x = (const float*)d_in[0];
  float* out = (float*)d_out;

  // 4 batches * (8 fwd + 4 bwd) workgroups of 128 threads (4 waves)
  knn_chain_kernel<<<B_ * 12, 128, 0, stream>>>(x, out);

  const size_t S1 = (size_t)B_ * T_ * N_ * K_;
  finalize_kernel<<<(unsigned)((S1 + 255) / 256), 256, 0, stream>>>(out);
}